// RGCN_graph_80753975099823
// MI455X (gfx1250) — compile-verified
//
#include <hip/hip_runtime.h>
#include <hip/hip_bf16.h>

// Problem constants (match reference)
#define NN   50000
#define EE   1600000
#define RR   12
#define FIN  64
#define HH   128
#define CC   10
#define GG   256
#define BNEPS 1e-5f

typedef __attribute__((ext_vector_type(2))) float v2f;
typedef __attribute__((ext_vector_type(8))) float v8f;

// ---------------------------------------------------------------------------
// Edge -> (dst,rel) segment counts
// ---------------------------------------------------------------------------
__global__ void edge_count_kernel(const int* __restrict__ dst,
                                  const int* __restrict__ etype,
                                  float* __restrict__ cnt) {
    long e = (long)blockIdx.x * blockDim.x + threadIdx.x;
    if (e >= EE) return;
    atomicAdd(&cnt[(long)dst[e] * RR + etype[e]], 1.0f);
}

// ---------------------------------------------------------------------------
// Scatter: hsum[dst*R+rel][0:F] += X[src][0:F]   (float4 per thread)
// ---------------------------------------------------------------------------
template<int F>
__global__ void scatter_kernel(const float* __restrict__ X,
                               const int* __restrict__ src,
                               const int* __restrict__ dst,
                               const int* __restrict__ etype,
                               float* __restrict__ hsum) {
    constexpr int QPE = F / 4;                     // quads per edge
    long gid = (long)blockIdx.x * blockDim.x + threadIdx.x;
    if (gid >= (long)EE * QPE) return;
    long e = gid / QPE;
    int  q = (int)(gid - e * QPE);
    int  s = src[e], d = dst[e], r = etype[e];
    float4 v = ((const float4*)(X + (long)s * F))[q];
    float* o = hsum + ((long)d * RR + r) * F + q * 4;
    atomicAdd(o + 0, v.x);
    atomicAdd(o + 1, v.y);
    atomicAdd(o + 2, v.z);
    atomicAdd(o + 3, v.w);
}

// ---------------------------------------------------------------------------
// hmean = hsum / max(cnt, 1)
// ---------------------------------------------------------------------------
template<int F>
__global__ void finalize_kernel(float* __restrict__ hsum,
                                const float* __restrict__ cnt) {
    long i = (long)blockIdx.x * blockDim.x + threadIdx.x;
    if (i >= (long)NN * RR * F) return;
    float c = cnt[i / F];
    hsum[i] *= (c > 1.0f) ? (1.0f / c) : 1.0f;
}

// ---------------------------------------------------------------------------
// WMMA GEMM:  Z[N,128] = [A0 | A1] @ [B0 ; B1] + bias
//   A0: [N,F0] (self features), A1: [N,KH] (per-relation means, KH = R*F)
//   B0: [F0,128] (root),        B1: [KH,128] (Wrel flattened)
// One wave computes a 16x128 output tile using v_wmma_f32_16x16x4_f32.
// ---------------------------------------------------------------------------
template<int F0, int KH>
__global__ __launch_bounds__(32)
void gemm_wmma_kernel(const float* __restrict__ A0,
                      const float* __restrict__ A1,
                      const float* __restrict__ B0,
                      const float* __restrict__ B1,
                      const float* __restrict__ bias,
                      float* __restrict__ Z) {
    constexpr int K = F0 + KH;
    const int tile  = blockIdx.x;          // M-tile (N == 3125 * 16 exactly)
    const int lane  = threadIdx.x;         // wave32
    const int lrow  = lane & 15;           // row-in-tile (A) / col-in-tile (B)
    const int khalf = lane >> 4;           // 0 -> K{0,1}, 1 -> K{2,3}

    const long arow = (long)tile * 16 + lrow;
    const float* a0 = A0 + arow * F0;
    const float* a1 = A1 + arow * KH;

    v8f acc[8] = {};                       // 16x128 f32 accumulators

    for (int kc = 0; kc < K; kc += 4) {
        const int k0 = kc + khalf * 2;
        v2f a;
        const float* Brow0;
        if (kc < F0) {                     // F0 % 4 == 0 -> quad fully in A0/B0
            a.x = a0[k0];
            a.y = a0[k0 + 1];
            Brow0 = B0 + (long)k0 * HH;
        } else {
            a.x = a1[k0 - F0];
            a.y = a1[k0 - F0 + 1];
            Brow0 = B1 + (long)(k0 - F0) * HH;
        }
        const float* Brow1 = Brow0 + HH;
#pragma unroll
        for (int nt = 0; nt < 8; ++nt) {
            v2f b;
            b.x = Brow0[nt * 16 + lrow];
            b.y = Brow1[nt * 16 + lrow];
            acc[nt] = __builtin_amdgcn_wmma_f32_16x16x4_f32(
                false, a, false, b, (short)0, acc[nt], false, false);
        }
    }

    // Epilogue: VGPR v, lanes 0-15 -> M=v, lanes 16-31 -> M=v+8
    const int rbase = tile * 16 + khalf * 8;
#pragma unroll
    for (int nt = 0; nt < 8; ++nt) {
        const int col = nt * 16 + lrow;
        const float bv = bias[col];
#pragma unroll
        for (int v = 0; v < 8; ++v) {
            Z[(long)(rbase + v) * HH + col] = acc[nt][v] + bv;
        }
    }
}

// ---------------------------------------------------------------------------
// BatchNorm statistics: one block per channel (128 blocks x 256 threads)
// ---------------------------------------------------------------------------
__global__ void bn_stats_kernel(const float* __restrict__ Z,
                                float* __restrict__ mu,
                                float* __restrict__ rsig) {
    __shared__ float sh[256];
    __shared__ float sh2[256];
    const int c = blockIdx.x;
    float s = 0.f, s2 = 0.f;
    for (int n = threadIdx.x; n < NN; n += blockDim.x) {
        float v = Z[(long)n * HH + c];
        s += v;
        s2 += v * v;
    }
    sh[threadIdx.x] = s;
    sh2[threadIdx.x] = s2;
    __syncthreads();
    for (int off = 128; off > 0; off >>= 1) {
        if (threadIdx.x < off) {
            sh[threadIdx.x]  += sh[threadIdx.x + off];
            sh2[threadIdx.x] += sh2[threadIdx.x + off];
        }
        __syncthreads();
    }
    if (threadIdx.x == 0) {
        float m   = sh[0] / (float)NN;
        float var = sh2[0] / (float)NN - m * m;
        mu[c]   = m;
        rsig[c] = rsqrtf(var + BNEPS);
    }
}

// ---------------------------------------------------------------------------
// BN apply + ReLU (in place)
// ---------------------------------------------------------------------------
__global__ void bn_apply_kernel(float* __restrict__ Z,
                                const float* __restrict__ gamma,
                                const float* __restrict__ beta,
                                const float* __restrict__ mu,
                                const float* __restrict__ rsig) {
    long i = (long)blockIdx.x * blockDim.x + threadIdx.x;
    if (i >= (long)NN * HH) return;
    int c = (int)(i & (HH - 1));
    float v = gamma[c] * (Z[i] - mu[c]) * rsig[c] + beta[c];
    Z[i] = v > 0.f ? v : 0.f;
}

// ---------------------------------------------------------------------------
// BN apply + ReLU + atomic mean-pool scatter
// ---------------------------------------------------------------------------
__global__ void bn_apply_pool_kernel(const float* __restrict__ Z,
                                     const float* __restrict__ gamma,
                                     const float* __restrict__ beta,
                                     const float* __restrict__ mu,
                                     const float* __restrict__ rsig,
                                     const int* __restrict__ batch,
                                     float* __restrict__ pooled) {
    long i = (long)blockIdx.x * blockDim.x + threadIdx.x;
    if (i >= (long)NN * HH) return;
    int  c = (int)(i & (HH - 1));
    long n = i >> 7;
    float v = gamma[c] * (Z[i] - mu[c]) * rsig[c] + beta[c];
    v = v > 0.f ? v : 0.f;
    atomicAdd(&pooled[(long)batch[n] * HH + c], v);
}

__global__ void graph_count_kernel(const int* __restrict__ batch,
                                   float* __restrict__ gcnt) {
    long i = (long)blockIdx.x * blockDim.x + threadIdx.x;
    if (i >= NN) return;
    atomicAdd(&gcnt[batch[i]], 1.0f);
}

// ---------------------------------------------------------------------------
// Final linear: out[G,C] = (pooled / max(gcnt,1)) @ Wf + bf
// ---------------------------------------------------------------------------
__global__ void final_linear_kernel(const float* __restrict__ pooled,
                                    const float* __restrict__ gcnt,
                                    const float* __restrict__ Wf,
                                    const float* __restrict__ bf,
                                    float* __restrict__ out) {
    int g = threadIdx.x;                   // 256 graphs, one block
    float inv = 1.0f / fmaxf(gcnt[g], 1.0f);
    float acc[CC];
#pragma unroll
    for (int c = 0; c < CC; ++c) acc[c] = bf[c];
    for (int k = 0; k < HH; ++k) {
        float p = pooled[(long)g * HH + k] * inv;
#pragma unroll
        for (int c = 0; c < CC; ++c) acc[c] += p * Wf[k * CC + c];
    }
#pragma unroll
    for (int c = 0; c < CC; ++c) out[(long)g * CC + c] = acc[c];
}

// ---------------------------------------------------------------------------
// Launcher
// ---------------------------------------------------------------------------
extern "C" void kernel_launch(void* const* d_in, const int* in_sizes, int n_in,
                              void* d_out, int out_size, void* d_ws, size_t ws_size,
                              hipStream_t stream) {
    const float* x      = (const float*)d_in[0];
    const float* Wrel1  = (const float*)d_in[1];
    const float* root1  = (const float*)d_in[2];
    const float* bias1  = (const float*)d_in[3];
    const float* gamma1 = (const float*)d_in[4];
    const float* beta1  = (const float*)d_in[5];
    const float* Wrel2  = (const float*)d_in[6];
    const float* root2  = (const float*)d_in[7];
    const float* bias2  = (const float*)d_in[8];
    const float* gamma2 = (const float*)d_in[9];
    const float* beta2  = (const float*)d_in[10];
    const float* Wf     = (const float*)d_in[11];
    const float* bf     = (const float*)d_in[12];
    const int* edge_index = (const int*)d_in[13];
    const int* etype      = (const int*)d_in[14];
    const int* batch      = (const int*)d_in[15];
    const int* src = edge_index;
    const int* dst = edge_index + EE;

    // Workspace layout (floats)
    float* ws     = (float*)d_ws;
    float* hsum   = ws;                                  // N*R*128 (max)
    float* cnt    = hsum + (size_t)NN * RR * HH;         // N*R
    float* bufA   = cnt  + (size_t)NN * RR;              // N*128 (z1 -> h1 in place)
    float* bufB   = bufA + (size_t)NN * HH;              // N*128 (z2)
    float* mu1    = bufB + (size_t)NN * HH;              // 128
    float* rs1    = mu1 + HH;                            // 128
    float* mu2    = rs1 + HH;                            // 128
    float* rs2    = mu2 + HH;                            // 128
    float* pooled = rs2 + HH;                            // G*128
    float* gcnt   = pooled + (size_t)GG * HH;            // G

    const int TB = 256;

    // ---- shared setup ----
    hipMemsetAsync(cnt, 0, (size_t)NN * RR * sizeof(float), stream);
    hipMemsetAsync(pooled, 0, ((size_t)GG * HH + GG) * sizeof(float), stream);
    edge_count_kernel<<<(EE + TB - 1) / TB, TB, 0, stream>>>(dst, etype, cnt);
    graph_count_kernel<<<(NN + TB - 1) / TB, TB, 0, stream>>>(batch, gcnt);

    // ---- layer 1 ----
    hipMemsetAsync(hsum, 0, (size_t)NN * RR * FIN * sizeof(float), stream);
    {
        long t = (long)EE * (FIN / 4);
        scatter_kernel<FIN><<<(unsigned)((t + TB - 1) / TB), TB, 0, stream>>>(
            x, src, dst, etype, hsum);
        long f = (long)NN * RR * FIN;
        finalize_kernel<FIN><<<(unsigned)((f + TB - 1) / TB), TB, 0, stream>>>(hsum, cnt);
    }
    gemm_wmma_kernel<FIN, RR * FIN><<<NN / 16, 32, 0, stream>>>(
        x, hsum, root1, Wrel1, bias1, bufA);
    bn_stats_kernel<<<HH, TB, 0, stream>>>(bufA, mu1, rs1);
    bn_apply_kernel<<<(unsigned)(((long)NN * HH + TB - 1) / TB), TB, 0, stream>>>(
        bufA, gamma1, beta1, mu1, rs1);

    // ---- layer 2 ----
    hipMemsetAsync(hsum, 0, (size_t)NN * RR * HH * sizeof(float), stream);
    {
        long t = (long)EE * (HH / 4);
        scatter_kernel<HH><<<(unsigned)((t + TB - 1) / TB), TB, 0, stream>>>(
            bufA, src, dst, etype, hsum);
        long f = (long)NN * RR * HH;
        finalize_kernel<HH><<<(unsigned)((f + TB - 1) / TB), TB, 0, stream>>>(hsum, cnt);
    }
    gemm_wmma_kernel<HH, RR * HH><<<NN / 16, 32, 0, stream>>>(
        bufA, hsum, root2, Wrel2, bias2, bufB);
    bn_stats_kernel<<<HH, TB, 0, stream>>>(bufB, mu2, rs2);
    bn_apply_pool_kernel<<<(unsigned)(((long)NN * HH + TB - 1) / TB), TB, 0, stream>>>(
        bufB, gamma2, beta2, mu2, rs2, batch, pooled);

    // ---- head ----
    final_linear_kernel<<<1, GG, 0, stream>>>(pooled, gcnt, Wf, bf, (float*)d_out);
}